// PositionalScaledDotProductAttention_42125039239481
// MI455X (gfx1250) — compile-verified
//
#include <hip/hip_runtime.h>

typedef __attribute__((ext_vector_type(2))) float v2f;
typedef __attribute__((ext_vector_type(4))) float v4f;
typedef __attribute__((ext_vector_type(8))) float v8f;

#define LSEQ   1024
#define DHEAD  64
#define NBH    32          // B*H = 4*8
#define NWAVE  4           // waves per workgroup
#define SPITCH 1028        // LDS row pitch (words): %64==4 -> conflict-free column scans, 16B aligned
#define QPITCH 68          // %64==4
#define TEMP_INV 0.125f    // 1/sqrt(64)
#define NEG_INF_F (-1e9f)

__global__ __launch_bounds__(128)
void psdpa_kernel(const float* __restrict__ qg, const float* __restrict__ kg,
                  const float* __restrict__ vg, const float* __restrict__ rprk,
                  const float* __restrict__ rprv,
                  float* __restrict__ outg, float* __restrict__ attng)
{
    __shared__ __align__(16) float S[16][SPITCH];    // scores -> probs for this query tile
    __shared__ __align__(16) float qs[16][QPITCH];   // raw q tile (16 x 64)
    __shared__ __align__(16) float obuf[16][QPITCH]; // WMMA P@V staging (16 x 64)
    __shared__ float rs[16];                         // per-row 1/sum

    const int tid  = threadIdx.x;
    const int wave = tid >> 5;         // 0..3
    const int lane = tid & 31;         // wave32
    const int bh   = blockIdx.x;       // fast dim -> L2 reuse of rpr slabs across b,h
    const int qt   = blockIdx.y;       // query tile (16 rows)
    const int q0   = qt * 16;
    const int lo   = lane & 15;        // N / column index in WMMA fragments
    const int hi   = lane >> 4;        // half-wave select
    const int k0   = hi * 2;           // K-pair base for f32 A/B fragments
    const int nvalid = 16 * (qt + 1);  // causal: keys [0, nvalid)

    // ---- load q tile into LDS (all 128 threads) -------------------------
    for (int i = tid; i < 16 * DHEAD; i += 128) {
        int m = i >> 6, d = i & 63;
        qs[m][d] = qg[((size_t)bh * LSEQ + q0 + m) * DHEAD + d];
    }
    __syncthreads();

    // ---- pre-scaled A fragments (per wave): A = q/T, 16 K-steps over D=64
    v2f af[16];
#pragma unroll
    for (int s = 0; s < 16; ++s) {
        v2f t = *(const v2f*)&qs[lo][4 * s + k0];
        af[s] = t * TEMP_INV;
    }

    // ===================== Phase 1: scores (kt strided over waves) =======
    for (int kt = wave; kt <= qt; kt += NWAVE) {
        v8f c;
        // rpr_k term (per-row GEMV, VALU) straight into WMMA C layout:
        // c[i] holds element (M = i + 8*hi, N = lo)
#pragma unroll 1
        for (int i = 0; i < 8; ++i) {
            const int m = i + 8 * hi;
            const float* rp = rprk + ((size_t)(q0 + m) * LSEQ + kt * 16 + lo) * DHEAD;
            float acc = 0.f;
#pragma unroll
            for (int d = 0; d < DHEAD; d += 4) {
                v4f r = *(const v4f*)(rp + d);
                acc = __builtin_fmaf(qs[m][d + 0], r.x, acc);
                acc = __builtin_fmaf(qs[m][d + 1], r.y, acc);
                acc = __builtin_fmaf(qs[m][d + 2], r.z, acc);
                acc = __builtin_fmaf(qs[m][d + 3], r.w, acc);
            }
            c[i] = acc;
        }
        // content term: += (q/T) @ k^T via 16 x V_WMMA_F32_16X16X4_F32
        const float* kb = kg + ((size_t)bh * LSEQ + kt * 16 + lo) * DHEAD + k0;
#pragma unroll
        for (int s = 0; s < 16; ++s) {
            v2f bf = *(const v2f*)(kb + 4 * s);  // B[kd][n] = K[n][kd]
            c = __builtin_amdgcn_wmma_f32_16x16x4_f32(
                    false, af[s], false, bf, (short)0, c, false, false);
        }
        if (kt == qt) {                          // causal mask, diagonal tile
#pragma unroll
            for (int i = 0; i < 8; ++i)
                if (lo > i + 8 * hi) c[i] = NEG_INF_F;
        }
#pragma unroll
        for (int i = 0; i < 8; ++i)
            S[i + 8 * hi][kt * 16 + lo] = c[i];
    }
    __syncthreads();

    // ---- softmax: 8 threads per row ------------------------------------
    {
        const int row = tid >> 3, sub = tid & 7;
        float mx = -3.4e38f;
        for (int j = sub; j < nvalid; j += 8) mx = fmaxf(mx, S[row][j]);
        mx = fmaxf(mx, __shfl_xor(mx, 1, 32));
        mx = fmaxf(mx, __shfl_xor(mx, 2, 32));
        mx = fmaxf(mx, __shfl_xor(mx, 4, 32));
        float sum = 0.f;
        for (int j = sub; j < nvalid; j += 8) {
            float e = __expf(S[row][j] - mx);
            S[row][j] = e;
            sum += e;
        }
        sum += __shfl_xor(sum, 1, 32);
        sum += __shfl_xor(sum, 2, 32);
        sum += __shfl_xor(sum, 4, 32);
        if (sub == 0) rs[row] = 1.f / sum;
    }
    __syncthreads();

    // ---- normalize in LDS + stream attn out (NT stores, rows per wave) --
    for (int m = wave; m < 16; m += NWAVE) {
        const float rv = rs[m];
        float* arow = attng + ((size_t)bh * LSEQ + q0 + m) * LSEQ;
        for (int j = 4 * lane; j < LSEQ; j += 128) {
            v4f p;
            if (j < nvalid) {
                p = *(v4f*)&S[m][j];
                p = p * rv;
                *(v4f*)&S[m][j] = p;                 // keep normalized probs
            } else {
                p.x = 0.f; p.y = 0.f; p.z = 0.f; p.w = 0.f;
            }
            __builtin_nontemporal_store(p, (v4f*)(arow + j));
        }
    }
    __syncthreads();

    // ===================== Phase 2: context ==============================
    // content part: out1 = P @ V; each wave owns one 16-wide d-tile (nt=wave)
    {
        const int nt = wave;
        v8f o;
#pragma unroll
        for (int i = 0; i < 8; ++i) o[i] = 0.f;
        for (int kt = 0; kt <= qt; ++kt) {
            const float* vb = vg + ((size_t)bh * LSEQ + kt * 16) * DHEAD + nt * 16 + lo;
#pragma unroll
            for (int s = 0; s < 4; ++s) {
                v2f pa = *(const v2f*)&S[lo][kt * 16 + 4 * s + k0];  // A = P k-slice
                v2f bf;
                bf.x = vb[(size_t)(4 * s + k0) * DHEAD];             // B[kd][n] = V[k][d]
                bf.y = vb[(size_t)(4 * s + k0 + 1) * DHEAD];
                o = __builtin_amdgcn_wmma_f32_16x16x4_f32(
                        false, pa, false, bf, (short)0, o, false, false);
            }
        }
#pragma unroll
        for (int i = 0; i < 8; ++i)
            obuf[i + 8 * hi][nt * 16 + lo] = o[i];
    }
    __syncthreads();

    // rpr_v part (rows per wave): contiguous float2 per lane per (q,k) row,
    // fused with the content add + final NT store.
    for (int m = wave; m < 16; m += NWAVE) {
        const float* rvp = rprv + (size_t)(q0 + m) * LSEQ * DHEAD + 2 * lane;
        float ax = 0.f, ay = 0.f;
#pragma unroll 4
        for (int j = 0; j < nvalid; ++j) {
            const float pr = S[m][j];                        // uniform LDS broadcast
            v2f rr = *(const v2f*)(rvp + (size_t)j * DHEAD);
            ax = __builtin_fmaf(pr, rr.x, ax);
            ay = __builtin_fmaf(pr, rr.y, ay);
        }
        v2f res;
        res.x = obuf[m][2 * lane]     + ax;
        res.y = obuf[m][2 * lane + 1] + ay;
        __builtin_nontemporal_store(res,
            (v2f*)(outg + ((size_t)bh * LSEQ + q0 + m) * DHEAD + 2 * lane));
    }
}

extern "C" void kernel_launch(void* const* d_in, const int* in_sizes, int n_in,
                              void* d_out, int out_size, void* d_ws, size_t ws_size,
                              hipStream_t stream) {
    const float* q    = (const float*)d_in[0];
    const float* k    = (const float*)d_in[1];
    const float* v    = (const float*)d_in[2];
    const float* rprk = (const float*)d_in[3];
    const float* rprv = (const float*)d_in[4];
    // d_in[5] = causal mask; known analytically, not read.
    float* out  = (float*)d_out;                        // [B,H,L,D]
    float* attn = out + (size_t)NBH * LSEQ * DHEAD;     // [B,H,L,L] follows in tuple order

    dim3 grid(NBH, LSEQ / 16);   // bh fastest -> concurrent blocks share rpr slab in L2
    psdpa_kernel<<<grid, 128, 0, stream>>>(q, k, v, rprk, rprv, out, attn);
}